// NodeTimesEdge_90512140796796
// MI455X (gfx1250) — compile-verified
//
#include <hip/hip_runtime.h>
#include <math.h>

#define D_IN  128
#define D_OUT 128

typedef __attribute__((ext_vector_type(2))) float v2f;
typedef __attribute__((ext_vector_type(8))) float v8f;

// ---------------------------------------------------------------------------
// Kernel 0: zero the scratch accumulators (agg[N*128] ++ deg[N], contiguous)
// ---------------------------------------------------------------------------
__global__ void gnn_zero_kernel(float* __restrict__ p, size_t n) {
    size_t i = (size_t)blockIdx.x * blockDim.x + threadIdx.x;
    if (i < n) p[i] = 0.0f;
}

// ---------------------------------------------------------------------------
// Kernel 1: edge scatter. One wave32 per edge: lane k owns 4 consecutive
// floats of the 128-wide row -> one b128 load + 4 f32 atomics (L2-resident:
// h (25.6MB) and agg (25.6MB) both fit in the 192MB L2).
// ---------------------------------------------------------------------------
__global__ void gnn_scatter_kernel(const float* __restrict__ h,
                                   const float* __restrict__ w,
                                   const int*   __restrict__ src,
                                   const int*   __restrict__ dst,
                                   float* __restrict__ agg,
                                   float* __restrict__ deg,
                                   int n_edges) {
    int gtid = blockIdx.x * blockDim.x + threadIdx.x;
    int e    = gtid >> 5;          // wave index == edge index
    int lane = threadIdx.x & 31;
    if (e >= n_edges) return;

    int   s  = src[e];
    int   d  = dst[e];
    float wt = w[e];

    const float4* hrow = (const float4*)(h + (size_t)s * D_IN);
    float4 v = hrow[lane];                       // global_load_b128

    float* arow = agg + (size_t)d * D_IN + lane * 4;
    atomicAdd(arow + 0, v.x * wt);
    atomicAdd(arow + 1, v.y * wt);
    atomicAdd(arow + 2, v.z * wt);
    atomicAdd(arow + 3, v.w * wt);

    if (lane == 0) atomicAdd(deg + d, 1.0f);
}

// ---------------------------------------------------------------------------
// Kernel 2: fused mean + tanh + GEMM (out = tanh(agg/deg) @ W^T + b).
// Block = 256 threads = 8 waves. Block computes one 16-node activated A tile
// into LDS; wave wv computes the 16x16 output tile for columns [wv*16, +16)
// using V_WMMA_F32_16X16X4_F32 over K=128 in steps of 4.
//
// Fragment layouts (ISA 7.12.2):
//   A 16x4 f32 : lane(0..15)=M, +16 selects K-half; VGPR j -> K = 2*half + j
//   B 4x16 f32 : lane(0..15)=N, +16 selects K-half; VGPR j -> K = 2*half + j
//   C/D 16x16  : lane l, VGPR r -> M = r + 8*(l>>4), N = l&15
// ---------------------------------------------------------------------------
__global__ void gnn_gemm_kernel(const float* __restrict__ agg,
                                const float* __restrict__ deg,
                                const float* __restrict__ Wt,   // [D_OUT][D_IN] row-major
                                const float* __restrict__ bias,
                                float* __restrict__ out,
                                int n_nodes) {
    __shared__ float Atile[16][D_IN + 4];   // +4 pad: kills LDS bank conflicts

    const int row0      = blockIdx.x * 16;           // first node of this M tile
    const int tid       = threadIdx.x;
    const bool full_tile = (row0 + 16 <= n_nodes);   // uniform: true for all tiles
                                                     // when n_nodes % 16 == 0

    // ---- stage 1: mean + tanh into LDS (2048 elems / 256 threads) ----
    if (full_tile) {
        #pragma unroll
        for (int i = tid; i < 16 * D_IN; i += 256) {
            int r = i >> 7, c = i & (D_IN - 1);
            int node = row0 + r;
            float dg = deg[node];
            float v  = (dg > 0.0f) ? tanhf(agg[(size_t)node * D_IN + c] / dg) : 0.0f;
            Atile[r][c] = v;
        }
    } else {
        for (int i = tid; i < 16 * D_IN; i += 256) {
            int r = i >> 7, c = i & (D_IN - 1);
            int node = row0 + r;
            float v = 0.0f;
            if (node < n_nodes) {
                float dg = deg[node];
                if (dg > 0.0f) v = tanhf(agg[(size_t)node * D_IN + c] / dg);
            }
            Atile[r][c] = v;
        }
    }
    __syncthreads();

    // ---- stage 2: WMMA f32 GEMM, one 16x16 tile per wave ----
    const int wv    = tid >> 5;        // N-tile index 0..7
    const int lane  = tid & 31;
    const int l15   = lane & 15;
    const int khalf = lane >> 4;       // 0: K={0,1}, 1: K={2,3} within a step
    const int ncol  = wv * 16 + l15;   // output feature this lane touches in B/D

    v8f acc;
    const float bv = bias[ncol];       // bias is constant down each column
    #pragma unroll
    for (int r = 0; r < 8; ++r) acc[r] = bv;

    const float* wrow = Wt + (size_t)ncol * D_IN;   // B[k][n] = W[n][k]

    #pragma unroll 4
    for (int k = 0; k < D_IN; k += 4) {
        v2f afrag, bfrag;
        afrag[0] = Atile[l15][k + khalf * 2 + 0];   // ds_load_2addr_b64 pairs
        afrag[1] = Atile[l15][k + khalf * 2 + 1];
        bfrag[0] = wrow[k + khalf * 2 + 0];         // global_load_b64 (L2/WGP$ hot)
        bfrag[1] = wrow[k + khalf * 2 + 1];
        acc = __builtin_amdgcn_wmma_f32_16x16x4_f32(
                  /*neg_a=*/false, afrag,
                  /*neg_b=*/false, bfrag,
                  /*c_mod=*/(short)0, acc,
                  /*reuse_a=*/false, /*reuse_b=*/false);
    }

    // ---- stage 3: store D tile ----
    const int mbase = row0 + khalf * 8;
    float* obase = out + (size_t)mbase * D_OUT + ncol;
    if (full_tile) {
        // 8 stores off one base pointer; r*D_OUT*4 = r*512 B fits in IOFFSET
        #pragma unroll
        for (int r = 0; r < 8; ++r)
            obase[(size_t)r * D_OUT] = acc[r];
    } else {
        #pragma unroll
        for (int r = 0; r < 8; ++r)
            if (mbase + r < n_nodes)
                obase[(size_t)r * D_OUT] = acc[r];
    }
}

// ---------------------------------------------------------------------------
extern "C" void kernel_launch(void* const* d_in, const int* in_sizes, int n_in,
                              void* d_out, int out_size, void* d_ws, size_t ws_size,
                              hipStream_t stream) {
    const float* h   = (const float*)d_in[0];
    const float* w   = (const float*)d_in[1];
    const int*   src = (const int*)  d_in[2];
    const int*   dst = (const int*)  d_in[3];
    const float* Wt  = (const float*)d_in[4];
    const float* b   = (const float*)d_in[5];
    float*       out = (float*)d_out;

    const int n_nodes = in_sizes[0] / D_IN;   // 50000
    const int n_edges = in_sizes[1];          // 800000

    float* agg = (float*)d_ws;                         // [n_nodes * 128]
    float* deg = agg + (size_t)n_nodes * D_IN;         // [n_nodes]

    // 0) zero accumulators (agg ++ deg are contiguous)
    size_t zn = (size_t)n_nodes * D_IN + (size_t)n_nodes;
    gnn_zero_kernel<<<(unsigned)((zn + 255) / 256), 256, 0, stream>>>(agg, zn);

    // 1) edge scatter: 8 waves (edges) per 256-thread block
    gnn_scatter_kernel<<<(n_edges + 7) / 8, 256, 0, stream>>>(
        h, w, src, dst, agg, deg, n_edges);

    // 2) fused mean+tanh+GEMM: one 16-node tile per block (50000 = 3125*16)
    gnn_gemm_kernel<<<(n_nodes + 15) / 16, 256, 0, stream>>>(
        agg, deg, Wt, b, out, n_nodes);
}